// PRNN_75033078661304
// MI455X (gfx1250) — compile-verified
//
#include <hip/hip_runtime.h>
#include <cmath>

typedef float v2f __attribute__((ext_vector_type(2)));
typedef float v4f __attribute__((ext_vector_type(4)));
typedef float v8f __attribute__((ext_vector_type(8)));

#define T_STEPS 128
#define LATS 48
#define LAT_STRIDE 20   // padded dword stride: 20 mod 4 == 0 (16B align), spreads banks

// J2 material constants (match the reference, folded at compile time)
constexpr float kNU      = 0.37f;
constexpr float kE       = 3130.0f;
constexpr float kMU      = kE / (2.0f * (1.0f + kNU));                       // shear modulus
constexpr float kLAM     = kE * kNU / ((1.0f + kNU) * (1.0f - 2.0f * kNU));  // Lame lambda
constexpr float kSIGY    = 64.8f;
constexpr float kH       = 300.0f;
constexpr float kTWO_MU  = 2.0f * kMU;
constexpr float kINV3MH  = 1.0f / (3.0f * kMU + kH);

__global__ __launch_bounds__(256) void prnn_j2_kernel(
    const float* __restrict__ x,   // [B, T, 3]
    const float* __restrict__ W1,  // [48, 3]
    const float* __restrict__ W2,  // [3, 48]
    float* __restrict__ out,       // [B, T, 3]
    int Btot)
{
    // Double-buffered LDS (parity on t) -> 3 barriers per timestep.
    __shared__ __align__(16) float xs[2][16][4];              // staged x_t per batch; [3] stays 0
    __shared__ __align__(16) float lat2[2][LATS][LAT_STRIDE]; // fc1 result, column-major [latent][batch]
    __shared__ __align__(16) float accw[2][16][4];            // fc2 accumulators per batch (3 used)
    __shared__ float w2s[3 * LATS];                           // softplus(W2)

    const int tid  = threadIdx.x;
    const int lane = tid & 31;
    const int wv   = tid >> 5;     // wave 0..7
    const int bb   = tid & 15;     // batch within block
    const int m    = tid >> 4;     // material point 0..15
    const int s15  = lane & 15;
    const bool hi  = lane >= 16;
    const int blockBatch = blockIdx.x * 16;

    // --- one-time prep -----------------------------------------------------
    if (tid < 3 * LATS) {
        float v = W2[tid];
        w2s[tid] = (v > 20.0f) ? v : log1pf(__expf(v));   // softplus
    }
    if (tid < 128) {
        ((float*)accw)[tid] = 0.0f;   // zero both acc buffers
        ((float*)xs)[tid]   = 0.0f;   // zero x staging incl. padding lane [3] (K=3 pad)
    }

    // fc1 B-fragments (K=4 x N=16, K=3 used): lanes 0-15 hold K0/K1, lanes 16-31 hold K2/K3(=0).
    // Wave wv owns latent columns [6*wv, 6*wv+5], covered by N-tiles n0..n1.
    const int n0 = (6 * wv) / 16;
    const int n1 = (6 * wv + 5) / 16;
    v2f bf0, bf1;
    {
        const int row0 = 16 * n0 + s15;
        bf0.x = hi ? W1[row0 * 3 + 2] : W1[row0 * 3 + 0];
        bf0.y = hi ? 0.0f            : W1[row0 * 3 + 1];
        const int row1 = 16 * n1 + s15;
        bf1.x = hi ? W1[row1 * 3 + 2] : W1[row1 * 3 + 0];
        bf1.y = hi ? 0.0f            : W1[row1 * 3 + 1];
    }

    __syncthreads();

    // per-lane fc2 weights for its material point
    const float w00 = w2s[0 * LATS + 3 * m + 0];
    const float w01 = w2s[0 * LATS + 3 * m + 1];
    const float w02 = w2s[0 * LATS + 3 * m + 2];
    const float w10 = w2s[1 * LATS + 3 * m + 0];
    const float w11 = w2s[1 * LATS + 3 * m + 1];
    const float w12 = w2s[1 * LATS + 3 * m + 2];
    const float w20 = w2s[2 * LATS + 3 * m + 0];
    const float w21 = w2s[2 * LATS + 3 * m + 1];
    const float w22 = w2s[2 * LATS + 3 * m + 2];

    // committed history: plastic strain tensor + accumulated plastic strain (registers)
    float p_xx = 0.0f, p_yy = 0.0f, p_zz = 0.0f, p_xy = 0.0f, alpha = 0.0f;

    // staged pointers (clamped so inactive lanes never form OOB addresses)
    const int tx = (tid < 16) ? tid : 0;
    const float* gxl = x + (size_t)(blockBatch + tx) * (T_STEPS * 3);
    const bool xActive = (tid < 16) && (blockBatch + tid < Btot);

    const int tq  = (tid < 48) ? tid : 0;
    const int bbq = tq / 3;
    const int oq  = tq - 3 * bbq;
    float* gout = out + (size_t)(blockBatch + bbq) * (T_STEPS * 3) + oq;
    const bool oActive = (tid < 48) && (blockBatch + bbq < Btot);

    // A-fragment source: one aligned b64 read per lane (K-pad read from the zeroed [3] slot)
    const int aoff = s15 * 4 + (hi ? 2 : 0);

    const int colLo = 6 * wv, colHi = 6 * wv + 5;
    const int c0 = 16 * n0 + s15;
    const int c1 = 16 * n1 + s15;
    const int mrow = hi ? 8 : 0;
    const bool store0 = (c0 >= colLo) && (c0 <= colHi);
    const bool store1 = (n1 != n0) && (c1 >= colLo) && (c1 <= colHi);

    const v8f cz = {0.f, 0.f, 0.f, 0.f, 0.f, 0.f, 0.f, 0.f};

    // --- time scan (unroll 2 -> LDS parity bases fold to immediates) --------
#pragma unroll 2
    for (int t = 0; t < T_STEPS; ++t) {
        const int p = t & 1;

        if (xActive) {
            const float* xp = gxl + t * 3;
            xs[p][tid][0] = xp[0];
            xs[p][tid][1] = xp[1];
            xs[p][tid][2] = xp[2];
            if (t + 1 < T_STEPS) __builtin_prefetch(xp + 3, 0, 1);
        }
        __syncthreads();  // A: x_t staged

        // A-fragment (M=16 batches x K=4): lanes 0-15 -> (x0,x1); lanes 16-31 -> (x2, 0)
        const v2f af = *(const v2f*)&xs[p][0][aoff];

        // fc1 via f32 WMMA; second op issued unconditionally (bf1==bf0 when n1==n0)
        const v8f d0 = __builtin_amdgcn_wmma_f32_16x16x4_f32(
            false, af, false, bf0, (short)0, cz, false, false);
        const v8f d1 = __builtin_amdgcn_wmma_f32_16x16x4_f32(
            false, af, false, bf1, (short)0, cz, false, false);

        // scatter my 6 latent columns into LDS (D layout: lane -> column, VGPR r -> row M)
        if (store0) {
            *(v4f*)&lat2[p][c0][mrow]     = (v4f){d0[0], d0[1], d0[2], d0[3]};
            *(v4f*)&lat2[p][c0][mrow + 4] = (v4f){d0[4], d0[5], d0[6], d0[7]};
        }
        if (store1) {
            *(v4f*)&lat2[p][c1][mrow]     = (v4f){d1[0], d1[1], d1[2], d1[3]};
            *(v4f*)&lat2[p][c1][mrow + 4] = (v4f){d1[4], d1[5], d1[6], d1[7]};
        }
        __syncthreads();  // B: latents ready

        // this lane's strain components
        const float e0 = lat2[p][3 * m + 0][bb];
        const float e1 = lat2[p][3 * m + 1][bb];
        const float e2 = lat2[p][3 * m + 2][bb];

        // --- J2 elastic predictor + radial return --------------------------
        const float exx = e0 - p_xx;
        const float eyy = e1 - p_yy;
        const float ezz = -p_zz;                 // plane strain
        const float exy = 0.5f * e2 - p_xy;      // tensorial shear
        const float tr  = exx + eyy + ezz;
        const float lt  = kLAM * tr;
        const float sxx = lt + kTWO_MU * exx;
        const float syy = lt + kTWO_MU * eyy;
        const float szz = lt + kTWO_MU * ezz;
        const float sxy = kTWO_MU * exy;
        const float pm  = (sxx + syy + szz) * (1.0f / 3.0f);
        const float dxx = sxx - pm, dyy = syy - pm, dzz = szz - pm;
        const float q   = sqrtf(1.5f * (dxx * dxx + dyy * dyy + dzz * dzz + 2.0f * sxy * sxy));
        const float fy  = q - (kSIGY + kH * alpha);
        const float dg  = fmaxf(fy, 0.0f) * kINV3MH;
        const float qs  = fmaxf(q, 1e-12f);
        const float sc  = 1.5f / qs;
        const float nxx = dxx * sc, nyy = dyy * sc, nzz = dzz * sc, nxy = sxy * sc;
        const float tmd = kTWO_MU * dg;
        const float oxx = sxx - tmd * nxx;
        const float oyy = syy - tmd * nyy;
        const float oxy = sxy - tmd * nxy;
        p_xx += dg * nxx;  p_yy += dg * nyy;  p_zz += dg * nzz;  p_xy += dg * nxy;
        alpha += dg;

        // --- fc2 partials + reduction over the 16 points of batch bb --------
        float s0 = oxx * w00 + oyy * w01 + oxy * w02;
        float s1 = oxx * w10 + oyy * w11 + oxy * w12;
        float s2 = oxx * w20 + oyy * w21 + oxy * w22;
        // lanes l and l^16 hold points 2*wv and 2*wv+1 of the same batch
        s0 += __shfl_xor(s0, 16, 32);
        s1 += __shfl_xor(s1, 16, 32);
        s2 += __shfl_xor(s2, 16, 32);
        if (!hi) {
            atomicAdd(&accw[p][bb][0], s0);
            atomicAdd(&accw[p][bb][1], s1);
            atomicAdd(&accw[p][bb][2], s2);
        }
        __syncthreads();  // C: accumulators complete

        if (oActive) {
            gout[t * 3] = accw[p][bbq][oq];
            accw[p][bbq][oq] = 0.0f;   // re-arm this parity for t+2
        }
    }
}

extern "C" void kernel_launch(void* const* d_in, const int* in_sizes, int n_in,
                              void* d_out, int out_size, void* d_ws, size_t ws_size,
                              hipStream_t stream) {
    (void)n_in; (void)out_size; (void)d_ws; (void)ws_size;
    const float* x  = (const float*)d_in[0];
    const float* W1 = (const float*)d_in[1];
    const float* W2 = (const float*)d_in[2];
    float* out = (float*)d_out;

    const int B = in_sizes[0] / (T_STEPS * 3);   // 16384
    const int grid = (B + 15) / 16;              // 16 batches (256 material points) per block
    prnn_j2_kernel<<<grid, 256, 0, stream>>>(x, W1, W2, out, B);
}